// GraphSAGE_12687333392404
// MI455X (gfx1250) — compile-verified
//
#include <hip/hip_runtime.h>

typedef float v2f __attribute__((ext_vector_type(2)));
typedef float v8f __attribute__((ext_vector_type(8)));
typedef unsigned int u32;
typedef u32 u32x4 __attribute__((ext_vector_type(4)));
typedef int  i32x4 __attribute__((ext_vector_type(4)));
typedef int  i32x8 __attribute__((ext_vector_type(8)));

// ---------------- degree count: cnt[dst] += 1 ----------------
__global__ __launch_bounds__(256)
void k_count(const int* __restrict__ dstI, float* __restrict__ cnt, int E) {
    int e = blockIdx.x * 256 + threadIdx.x;
    if (e < E) atomicAdd(&cnt[dstI[e]], 1.0f);
}

// ---------------- cnt -> 1/max(cnt,1) in place ----------------
__global__ __launch_bounds__(256)
void k_invert(float* __restrict__ cnt, int n) {
    int i = blockIdx.x * 256 + threadIdx.x;
    if (i < n) cnt[i] = 1.0f / fmaxf(cnt[i], 1.0f);
}

// ---------------- scatter-add: summed[dst] += feats[src], F = 1<<Fshift ----------------
__global__ __launch_bounds__(256)
void k_scatter(const float* __restrict__ feats, const int* __restrict__ srcI,
               const int* __restrict__ dstI, float* __restrict__ summed,
               int E, int Fshift) {
    unsigned id = blockIdx.x * 256u + threadIdx.x;
    unsigned total = (unsigned)E << Fshift;
    if (id >= total) return;
    int e = (int)(id >> Fshift);
    int f = (int)(id & ((1u << Fshift) - 1u));
    atomicAdd(&summed[((long long)dstI[e] << Fshift) + f],
              feats[((long long)srcI[e] << Fshift) + f]);
}

// ---------------- TDM: DMA a [rows x rowlen] f32 strip (contiguous rows) into LDS ----------------
// LDS rows are padded by 4 dwords (pad_interval = one row) to kill bank conflicts.
// intervalCode: 2^(code+1) dwords per pad interval -> 6 for 128-dword rows, 7 for 256.
__device__ __forceinline__ void tdm_load_strip(u32 lds_addr, const float* gptr,
                                               u32 rows, u32 rowlen, u32 intervalCode) {
    unsigned long long ga = (unsigned long long)(uintptr_t)gptr;
    u32x4 g0;
    g0[0] = 1u;                                              // count=1, load, no gather
    g0[1] = lds_addr;                                        // LDS byte address
    g0[2] = (u32)(ga & 0xFFFFFFFFu);                         // global_addr[31:0]
    g0[3] = (u32)((ga >> 32) & 0x01FFFFFFu) | 0x80000000u;   // global_addr[56:32] | type=2
    i32x8 g1;
    g1[0] = (int)((2u << 16)            // data_size = 4 bytes
                | (1u << 20)            // pad_enable
                | (intervalCode << 22)  // pad_interval = one row
                | (3u << 25));          // pad_amount = 4 dwords
    g1[1] = (int)((rowlen & 0xFFFFu) << 16);                 // tensor_dim0[15:0]
    g1[2] = (int)(((rowlen >> 16) & 0xFFFFu)
                | ((rows & 0xFFFFu) << 16));                 // tensor_dim0[31:16] | tensor_dim1[15:0]
    g1[3] = (int)(((rows >> 16) & 0xFFFFu)
                | ((rowlen & 0xFFFFu) << 16));               // tensor_dim1[31:16] | tile_dim0
    g1[4] = (int)(rows & 0xFFFFu);                           // tile_dim1 | tile_dim2=0
    g1[5] = (int)rowlen;                                     // tensor_dim0_stride[31:0]
    g1[6] = 0;
    g1[7] = 0;
    i32x4 z4 = {0, 0, 0, 0};                                 // 2-D tile: groups 2/3 unused
    i32x8 z8 = {0, 0, 0, 0, 0, 0, 0, 0};
    __builtin_amdgcn_tensor_load_to_lds(g0, g1, z4, z4, z8, 0);
}

// ---------------- fused SAGE layer GEMM ----------------
// out[m][n] = inv[m]*sum_k Asum[m][k]*Wl[n][k] + sum_k Ax[m][k]*Wr[n][k] + bias[n]
// N fixed = 256. Each wave computes a 16x64 strip via V_WMMA_F32_16X16X4_F32.
// Weight strips (64 x K each for Wl, Wr) are TDM-staged into LDS once per block.
__global__ __launch_bounds__(256)
void k_sage_gemm(const float* __restrict__ Asum, const float* __restrict__ Ax,
                 const float* __restrict__ inv,
                 const float* __restrict__ Wl, const float* __restrict__ Wr,
                 const float* __restrict__ bias, float* __restrict__ out,
                 int M, int K, int do_relu) {
    extern __shared__ float smem[];                  // 2 * 64 * (K+4) floats
    const int lane = threadIdx.x & 31;
    const int wave = threadIdx.x >> 5;
    const int mt   = blockIdx.x * 8 + wave;          // 16-row tile index
    const int n0   = blockIdx.y << 6;                // 64-col macro tile
    const int strideW = K + 4;                       // padded LDS row stride (dwords)

    // ---- stage Wl/Wr strips [n0..n0+63] x K into LDS via Tensor Data Mover ----
    if (wave == 0) {
        const u32 icode = (K == 128) ? 6u : 7u;      // pad every K dwords
        const u32 base  = (u32)(uintptr_t)smem;
        const u32 tile1 = (u32)(64u * (u32)strideW * 4u);
        tdm_load_strip(base,         Wl + (size_t)n0 * K, 64u, (u32)K, icode);
        tdm_load_strip(base + tile1, Wr + (size_t)n0 * K, 64u, (u32)K, icode);
        __builtin_amdgcn_s_wait_tensorcnt(0);
    }
    __syncthreads();

    if ((mt << 4) >= M) return;                      // wave-uniform exit after barrier

    const int ml = lane & 15;                        // m (A/C) and n (B/C) lane index
    const int h2 = (lane >> 4) << 1;                 // k sub-offset 0 or 2
    const int m  = (mt << 4) + ml;
    const float s = inv[m];

    v8f c[4];
#pragma unroll
    for (int j = 0; j < 4; ++j) {
        const float bj = bias[n0 + (j << 4) + ml];
#pragma unroll
        for (int v = 0; v < 8; ++v) c[j][v] = bj;
    }

    const float* aSrow = Asum + (long long)m * K + h2;
    const float* aXrow = Ax   + (long long)m * K + h2;
    const float* wlT   = smem;                       // [64][K+4]
    const float* wrT   = smem + 64 * strideW;

    for (int k0 = 0; k0 < K; k0 += 4) {
        v2f aS = *(const v2f*)(aSrow + k0);
        aS = aS * s;                                 // fused mean (divide by degree)
        v2f aX = *(const v2f*)(aXrow + k0);
#pragma unroll
        for (int j = 0; j < 4; ++j) {
            const int nL = (j << 4) + ml;            // local weight row in LDS
            v2f bL = *(const v2f*)(wlT + nL * strideW + k0 + h2);
            v2f bR = *(const v2f*)(wrT + nL * strideW + k0 + h2);
            c[j] = __builtin_amdgcn_wmma_f32_16x16x4_f32(
                false, aS, false, bL, (short)0, c[j], false, false);
            c[j] = __builtin_amdgcn_wmma_f32_16x16x4_f32(
                false, aX, false, bR, (short)0, c[j], false, false);
        }
    }

    const int rbase = (mt << 4) + ((lane >> 4) << 3); // C: m = vgpr + (lane/16)*8
#pragma unroll
    for (int j = 0; j < 4; ++j) {
        const int col = n0 + (j << 4) + ml;
#pragma unroll
        for (int v = 0; v < 8; ++v) {
            float val = c[j][v];
            if (do_relu) val = fmaxf(val, 0.0f);
            out[(long long)(rbase + v) * 256 + col] = val;
        }
    }
}

// ---------------- heads: out1 = h@Wh1.T + bh1 (4), out2 = h@Wh2.T + bh2 (3) ----------------
__global__ __launch_bounds__(256)
void k_heads(const float* __restrict__ h, const float* __restrict__ Wh1,
             const float* __restrict__ bh1, const float* __restrict__ Wh2,
             const float* __restrict__ bh2, float* __restrict__ out1,
             float* __restrict__ out2, int M) {
    int i = blockIdx.x * 256 + threadIdx.x;
    if (i >= M) return;
    float a1[4] = {0.f, 0.f, 0.f, 0.f};
    float a2[3] = {0.f, 0.f, 0.f};
    const float4* hrow = (const float4*)(h + (long long)i * 256);
    for (int k = 0; k < 64; ++k) {
        float4 hv = hrow[k];
#pragma unroll
        for (int o = 0; o < 4; ++o) {
            float4 w = ((const float4*)(Wh1 + o * 256))[k];
            a1[o] += hv.x * w.x + hv.y * w.y + hv.z * w.z + hv.w * w.w;
        }
#pragma unroll
        for (int o = 0; o < 3; ++o) {
            float4 w = ((const float4*)(Wh2 + o * 256))[k];
            a2[o] += hv.x * w.x + hv.y * w.y + hv.z * w.z + hv.w * w.w;
        }
    }
#pragma unroll
    for (int o = 0; o < 4; ++o) out1[(long long)i * 4 + o] = a1[o] + bh1[o];
#pragma unroll
    for (int o = 0; o < 3; ++o) out2[(long long)i * 3 + o] = a2[o] + bh2[o];
}

extern "C" void kernel_launch(void* const* d_in, const int* in_sizes, int n_in,
                              void* d_out, int out_size, void* d_ws, size_t ws_size,
                              hipStream_t stream) {
    const float* x    = (const float*)d_in[0];
    const int*   ei   = (const int*)d_in[1];
    const float* W1l  = (const float*)d_in[2];
    const float* b1l  = (const float*)d_in[3];
    const float* W1r  = (const float*)d_in[4];
    const float* W2l  = (const float*)d_in[5];
    const float* b2l  = (const float*)d_in[6];
    const float* W2r  = (const float*)d_in[7];
    const float* Wh1  = (const float*)d_in[8];
    const float* bh1  = (const float*)d_in[9];
    const float* Wh2  = (const float*)d_in[10];
    const float* bh2  = (const float*)d_in[11];

    const int M = in_sizes[0] / 128;   // 50000 (divisible by 16)
    const int E = in_sizes[1] / 2;     // 800000
    const int* srcI = ei;              // edge_index[0]
    const int* dstI = ei + E;          // edge_index[1]

    // workspace layout (floats): inv[M] | summed[M*256] | h1[M*256]
    float* ws = (float*)d_ws;
    size_t invPad = ((size_t)M + 127) & ~(size_t)127;
    float* inv    = ws;
    float* summed = ws + invPad;
    float* h1     = summed + (size_t)M * 256;

    // output layout (floats): out1[M*4] | out2[M*3] | h[M*256]
    float* out1 = (float*)d_out;
    float* out2 = out1 + (size_t)M * 4;
    float* hout = out1 + (size_t)M * 7;

    // ---- degrees (shared by both layers) ----
    (void)hipMemsetAsync(inv, 0, (size_t)M * sizeof(float), stream);
    k_count<<<(E + 255) / 256, 256, 0, stream>>>(dstI, inv, E);
    k_invert<<<(M + 255) / 256, 256, 0, stream>>>(inv, M);

    // ---- layer 1: mean-agg (F=128) + fused GEMM + ReLU ----
    (void)hipMemsetAsync(summed, 0, (size_t)M * 128 * sizeof(float), stream);
    {
        unsigned total = (unsigned)E << 7;
        k_scatter<<<(total + 255) / 256, 256, 0, stream>>>(x, srcI, dstI, summed, E, 7);
    }
    dim3 gGemm((M / 16 + 7) / 8, 4);   // 8 waves/block * 16 rows, 4 x 64-col macro tiles
    {
        size_t lds1 = 2u * 64u * (128u + 4u) * 4u;   // 67.5 KB
        k_sage_gemm<<<gGemm, 256, lds1, stream>>>(summed, x, inv, W1l, W1r, b1l, h1, M, 128, 1);
    }

    // ---- layer 2: mean-agg (F=256) + fused GEMM (h -> d_out slice) ----
    (void)hipMemsetAsync(summed, 0, (size_t)M * 256 * sizeof(float), stream);
    {
        unsigned total = (unsigned)E << 8;
        k_scatter<<<(total + 255) / 256, 256, 0, stream>>>(h1, srcI, dstI, summed, E, 8);
    }
    {
        size_t lds2 = 2u * 64u * (256u + 4u) * 4u;   // 130 KB
        k_sage_gemm<<<gGemm, 256, lds2, stream>>>(summed, h1, inv, W2l, W2r, b2l, hout, M, 256, 0);
    }

    // ---- classifier heads ----
    k_heads<<<(M + 255) / 256, 256, 0, stream>>>(hout, Wh1, bh1, Wh2, bh2, out1, out2, M);
}